// GraphConv_7937099563613
// MI455X (gfx1250) — compile-verified
//
#include <hip/hip_runtime.h>
#include <hip/hip_bf16.h>

typedef __attribute__((ext_vector_type(2))) float v2f;
typedef __attribute__((ext_vector_type(8))) float v8f;

#define D 64  // D_IN == D_OUT == 64

// ---------------------------------------------------------------------------
// Kernel 1: per-node precompute with f32 WMMA.
//   A = x @ (W_top - W_bot) + bias      (used via dst gather)
//   B = x @ W_bot                       (used via src gather)
// One wave per 16-row tile; 4 N-tiles of 16 cover D_OUT=64; K loop in steps
// of 4 using V_WMMA_F32_16X16X4_F32.
// ---------------------------------------------------------------------------
__global__ __launch_bounds__(256) void edgeconv_gemm(
    const float* __restrict__ x, const float* __restrict__ W,
    const float* __restrict__ bias,
    float* __restrict__ A, float* __restrict__ Bm, int N)
{
    const int lane  = threadIdx.x & 31;
    const int wave  = threadIdx.x >> 5;
    const int m0    = (blockIdx.x * 8 + wave) * 16;   // 8 waves/block, 16 rows/wave
    const int mrow  = lane & 15;
    const int khalf = (lane >> 4) * 2;                // lanes 16-31 hold K+2,K+3

    int rowL = m0 + mrow;                             // clamp loads, guard stores
    if (rowL > N - 1) rowL = N - 1;
    const float* xrow = x + (size_t)rowL * D;

    v8f accA[4], accB[4];
#pragma unroll
    for (int t = 0; t < 4; ++t) {
        accA[t] = (v8f)(0.0f);
        accB[t] = (v8f)(0.0f);
    }

#pragma unroll
    for (int k = 0; k < D; k += 4) {
        // A-matrix fragment (16x4 f32): lane holds x[m][k+khalf .. k+khalf+1]
        v2f a;
        a.x = xrow[k + khalf];
        a.y = xrow[k + khalf + 1];
#pragma unroll
        for (int t = 0; t < 4; ++t) {
            const int n = t * 16 + (lane & 15);
            // B-matrix fragments (4x16 f32): lane holds W[k+khalf+v][n]
            const float wt0 = W[(size_t)(k + khalf)     * D + n];      // W_top row
            const float wt1 = W[(size_t)(k + khalf + 1) * D + n];
            const float wb0 = W[(size_t)(k + khalf + 64) * D + n];     // W_bot row
            const float wb1 = W[(size_t)(k + khalf + 65) * D + n];
            v2f b1, b2;
            b1.x = wt0 - wb0;  b1.y = wt1 - wb1;   // W_top - W_bot
            b2.x = wb0;        b2.y = wb1;         // W_bot
            accA[t] = __builtin_amdgcn_wmma_f32_16x16x4_f32(
                false, a, false, b1, (short)0, accA[t], false, false);
            accB[t] = __builtin_amdgcn_wmma_f32_16x16x4_f32(
                false, a, false, b2, (short)0, accB[t], false, false);
        }
    }

    // C/D layout: VGPR r -> M = r (lanes 0-15) / r+8 (lanes 16-31); N = lane&15
    const int rbase = (lane >> 4) * 8;
#pragma unroll
    for (int t = 0; t < 4; ++t) {
        const int col = t * 16 + (lane & 15);
        const float bv = bias[col];
#pragma unroll
        for (int r = 0; r < 8; ++r) {
            const int row = m0 + rbase + r;
            if (row < N) {
                A [(size_t)row * D + col] = accA[t][r] + bv;
                Bm[(size_t)row * D + col] = accB[t][r];
            }
        }
    }
}

// ---------------------------------------------------------------------------
// Kernel 2: per-edge gather + relu + atomic scatter-add.
// One wave per edge; each lane handles 2 features (float2). Edge indices are
// lane-uniform -> readfirstlane so addressing is SGPR-based.
// ---------------------------------------------------------------------------
__global__ __launch_bounds__(256) void edgeconv_edges(
    const int* __restrict__ ei, const float* __restrict__ A,
    const float* __restrict__ Bm, float* __restrict__ out,
    float* __restrict__ count, int E)
{
    const int e    = (int)((blockIdx.x * (unsigned)blockDim.x + threadIdx.x) >> 5);
    const int lane = threadIdx.x & 31;
    if (e >= E) return;                               // wave-uniform branch

    const int src = __builtin_amdgcn_readfirstlane(ei[e]);       // j
    const int dst = __builtin_amdgcn_readfirstlane(ei[E + e]);   // i (aggregate)

    const float2 av = *(const float2*)(A  + (size_t)dst * D + lane * 2);
    const float2 bv = *(const float2*)(Bm + (size_t)src * D + lane * 2);
    const float h0 = fmaxf(av.x + bv.x, 0.0f);
    const float h1 = fmaxf(av.y + bv.y, 0.0f);

    float* o = out + (size_t)dst * D + lane * 2;
    atomicAdd(o,     h0);
    atomicAdd(o + 1, h1);
    if (lane == 0) atomicAdd(count + dst, 1.0f);
}

// ---------------------------------------------------------------------------
// Kernel 3: out = sum / max(count, 1)
// ---------------------------------------------------------------------------
__global__ __launch_bounds__(256) void edgeconv_final(
    float* __restrict__ out, const float* __restrict__ count, int total)
{
    const int i = blockIdx.x * blockDim.x + threadIdx.x;
    if (i < total) {
        const float c = count[i >> 6];
        out[i] = out[i] / fmaxf(c, 1.0f);
    }
}

extern "C" void kernel_launch(void* const* d_in, const int* in_sizes, int n_in,
                              void* d_out, int out_size, void* d_ws, size_t ws_size,
                              hipStream_t stream)
{
    const float* x  = (const float*)d_in[0];
    const int*   ei = (const int*)  d_in[1];   // [2, E] row-major
    const float* W  = (const float*)d_in[2];   // [128, 64]
    const float* b  = (const float*)d_in[3];   // [64]

    const int N = in_sizes[0] / D;
    const int E = in_sizes[1] / 2;

    float* out   = (float*)d_out;
    float* A     = (float*)d_ws;               // [N, 64]
    float* Bm    = A  + (size_t)N * D;         // [N, 64]
    float* count = Bm + (size_t)N * D;         // [N]

    // Accumulator + count must start at zero every call (capture-safe memsets).
    hipMemsetAsync(d_out, 0, (size_t)out_size * sizeof(float), stream);
    hipMemsetAsync(count, 0, (size_t)N * sizeof(float), stream);

    const int gemmBlocks = (N + 127) / 128;            // 128 rows per block
    edgeconv_gemm<<<gemmBlocks, 256, 0, stream>>>(x, W, b, A, Bm, N);

    const int edgeBlocks = (E + 7) / 8;                // 8 edges (waves) per block
    edgeconv_edges<<<edgeBlocks, 256, 0, stream>>>(ei, A, Bm, out, count, E);

    const int finBlocks = (N * D + 255) / 256;
    edgeconv_final<<<finBlocks, 256, 0, stream>>>(out, count, N * D);
}